// LightGCN_72550587564378
// MI455X (gfx1250) — compile-verified
//
#include <hip/hip_runtime.h>
#include <hip/hip_bf16.h>
#include <math.h>

#define N_USERS_C 100000
#define N_ITEMS_C 50000
#define N_NODES_C (N_USERS_C + N_ITEMS_C)
#define EMB_DIM_C 64
#define NNZ_C     2000000
#define BATCH_C   16384
#define REG_C     1.0e-4f

typedef float v2f __attribute__((ext_vector_type(2)));
typedef float v8f __attribute__((ext_vector_type(8)));

// ---------------------------------------------------------------------------
// 1) Init: embA = acc = concat(user_emb, item_emb); counts = 0
// ---------------------------------------------------------------------------
__global__ void lg_init(const float* __restrict__ user_emb,
                        const float* __restrict__ item_emb,
                        float* __restrict__ embA, float* __restrict__ acc,
                        int* __restrict__ counts) {
  int i = blockIdx.x * blockDim.x + threadIdx.x;
  const int total = N_NODES_C * EMB_DIM_C;
  if (i < total) {
    int node = i >> 6;
    int d = i & 63;
    float v = (node < N_USERS_C)
                  ? user_emb[(size_t)node * EMB_DIM_C + d]
                  : item_emb[(size_t)(node - N_USERS_C) * EMB_DIM_C + d];
    embA[i] = v;
    acc[i] = v;
  }
  if (i < N_NODES_C) counts[i] = 0;
}

// ---------------------------------------------------------------------------
// 2) Histogram of row indices
// ---------------------------------------------------------------------------
__global__ void lg_count(const int* __restrict__ rows, int* __restrict__ counts) {
  int i = blockIdx.x * blockDim.x + threadIdx.x;
  int stride = gridDim.x * blockDim.x;
  for (; i < NNZ_C; i += stride) atomicAdd(&counts[rows[i]], 1);
}

// ---------------------------------------------------------------------------
// 3) Exclusive prefix sum over 150000 counts (single block, chunked scan)
//    row_start[0..N] (exclusive, row_start[N]=NNZ), cursor = copy of starts
// ---------------------------------------------------------------------------
__global__ void lg_scan(const int* __restrict__ counts,
                        int* __restrict__ row_start, int* __restrict__ cursor) {
  __shared__ int sdata[1024];
  __shared__ int s_carry;
  int tid = threadIdx.x;
  if (tid == 0) { s_carry = 0; row_start[0] = 0; }
  __syncthreads();
  for (int base = 0; base < N_NODES_C; base += 1024) {
    int i = base + tid;
    int v = (i < N_NODES_C) ? counts[i] : 0;
    sdata[tid] = v;
    __syncthreads();
    for (int off = 1; off < 1024; off <<= 1) {
      int t = (tid >= off) ? sdata[tid - off] : 0;
      __syncthreads();
      sdata[tid] += t;
      __syncthreads();
    }
    int incl = sdata[tid] + s_carry;
    if (i < N_NODES_C) {
      row_start[i + 1] = incl;   // inclusive -> start of row i+1
      cursor[i] = incl - v;      // exclusive -> start of row i
    }
    __syncthreads();
    if (tid == 1023) s_carry = incl;
    __syncthreads();
  }
}

// ---------------------------------------------------------------------------
// 4) Scatter edges into CSR order (col + val reordered for linear reads)
// ---------------------------------------------------------------------------
__global__ void lg_scatter(const int* __restrict__ rows, const int* __restrict__ cols,
                           const float* __restrict__ vals, int* __restrict__ cursor,
                           int* __restrict__ csr_col, float* __restrict__ csr_val) {
  int i = blockIdx.x * blockDim.x + threadIdx.x;
  int stride = gridDim.x * blockDim.x;
  for (; i < NNZ_C; i += stride) {
    int p = atomicAdd(&cursor[rows[i]], 1);
    csr_col[p] = cols[i];
    csr_val[p] = vals[i];
  }
}

// ---------------------------------------------------------------------------
// 5) SpMM layer: one wave32 per row; lanes cover 64 dims as float2.
//    Pure gather (no atomics); L2-resident x; prefetch next row's data.
//    Fused: y[r] = rowsum;  acc[r] += rowsum.
// ---------------------------------------------------------------------------
__global__ void __launch_bounds__(256) lg_spmm(
    const float* __restrict__ x, const int* __restrict__ row_start,
    const int* __restrict__ csr_col, const float* __restrict__ csr_val,
    float* __restrict__ y, float* __restrict__ acc) {
  int wid = (blockIdx.x * blockDim.x + threadIdx.x) >> 5;  // row id
  int lane = threadIdx.x & 31;
  if (wid >= N_NODES_C) return;
  int s = row_start[wid];
  int e = row_start[wid + 1];
  int d = lane << 1;
  float sx = 0.f, sy = 0.f;
  int c = (s < e) ? csr_col[s] : 0;
  for (int p = s; p < e; ++p) {
    float v = csr_val[p];
    int cn = (p + 1 < e) ? csr_col[p + 1] : c;
    __builtin_prefetch(x + (size_t)cn * EMB_DIM_C + d, 0, 1);  // global_prefetch_b8
    v2f xv = *(const v2f*)(x + (size_t)c * EMB_DIM_C + d);
    sx = fmaf(v, xv.x, sx);
    sy = fmaf(v, xv.y, sy);
    c = cn;
  }
  size_t o = (size_t)wid * EMB_DIM_C + d;
  v2f out; out.x = sx; out.y = sy;
  *(v2f*)(y + o) = out;
  v2f a = *(const v2f*)(acc + o);
  a.x += sx; a.y += sy;
  *(v2f*)(acc + o) = a;
}

// ---------------------------------------------------------------------------
// 6) BPR scoring via V_WMMA_F32_16X16X4_F32.
//    One wave handles 16 batch triplets. A = users tile (16x64), B = pos/neg
//    tile transposed (64x16). Accumulate over 16 K-steps of 4; scores are the
//    diagonal of the 16x16 result. 0.25 (light_out = acc/4) folded into loads.
//    Per-lane load pattern per ISA 7.12.2 (32-bit A 16x4 / B 4x16):
//      lane L: row/col = L&15, K-pair = k0 + 2*(L>>4).
//    Reg term: each element of the gathered tiles loaded exactly once per wave.
// ---------------------------------------------------------------------------
__global__ void __launch_bounds__(256) lg_score(
    const float* __restrict__ acc, const int* __restrict__ users,
    const int* __restrict__ pos, const int* __restrict__ neg,
    float* __restrict__ partials) {
  int wid = (blockIdx.x * blockDim.x + threadIdx.x) >> 5;
  int lane = threadIdx.x & 31;
  int base = wid << 4;
  int m = lane & 15;
  int khalf = (lane >> 4) << 1;  // 0 or 2
  int un = users[base + m];
  int pn = pos[base + m] + N_USERS_C;
  int nn = neg[base + m] + N_USERS_C;
  const float* urow = acc + (size_t)un * EMB_DIM_C + khalf;
  const float* prow = acc + (size_t)pn * EMB_DIM_C + khalf;
  const float* nrow = acc + (size_t)nn * EMB_DIM_C + khalf;

  v8f cpos = {};
  v8f cneg = {};
  float sq = 0.f;
#pragma unroll
  for (int k0 = 0; k0 < EMB_DIM_C; k0 += 4) {
    v2f a  = 0.25f * *(const v2f*)(urow + k0);
    v2f bp = 0.25f * *(const v2f*)(prow + k0);
    v2f bn = 0.25f * *(const v2f*)(nrow + k0);
    sq += a.x * a.x + a.y * a.y + bp.x * bp.x + bp.y * bp.y + bn.x * bn.x + bn.y * bn.y;
    cpos = __builtin_amdgcn_wmma_f32_16x16x4_f32(false, a, false, bp, (short)0, cpos,
                                                 false, false);
    cneg = __builtin_amdgcn_wmma_f32_16x16x4_f32(false, a, false, bn, (short)0, cneg,
                                                 false, false);
  }

  // Diagonal extraction: (j,j) lives at VGPR j, lane j (j<8) or VGPR j-8,
  // lane j+16 (j>=8)  -> active lanes 0..7 and 24..31.
  float ps = 0.f, ns = 0.f;
  bool active = false;
#pragma unroll
  for (int v = 0; v < 8; ++v) {
    bool dsel = (lane == v) || (lane == v + 24);
    if (dsel) { ps = cpos[v]; ns = cneg[v]; active = true; }
  }
  float contrib = 0.f;
  if (active) {
    float xv = ps - ns;
    // -log_sigmoid(x) = softplus(-x) = max(-x,0) + log1p(exp(-|x|))
    contrib = fmaxf(-xv, 0.f) + log1pf(expf(-fabsf(xv)));
  }
  float tot = contrib + (0.5f * REG_C) * sq;
#pragma unroll
  for (int off = 16; off > 0; off >>= 1) tot += __shfl_xor(tot, off, 32);
  if (lane == 0) partials[wid] = tot * (1.0f / (float)BATCH_C);
}

// ---------------------------------------------------------------------------
// 7) Deterministic final reduce of 1024 wave partials -> d_out[0]
// ---------------------------------------------------------------------------
__global__ void lg_reduce(const float* __restrict__ partials, float* __restrict__ out) {
  __shared__ float sdata[256];
  int tid = threadIdx.x;
  float s = 0.f;
  for (int i = tid; i < BATCH_C / 16; i += 256) s += partials[i];
  sdata[tid] = s;
  __syncthreads();
  for (int off = 128; off > 0; off >>= 1) {
    if (tid < off) sdata[tid] += sdata[tid + off];
    __syncthreads();
  }
  if (tid == 0) out[0] = sdata[0];
}

// ---------------------------------------------------------------------------
extern "C" void kernel_launch(void* const* d_in, const int* in_sizes, int n_in,
                              void* d_out, int out_size, void* d_ws, size_t ws_size,
                              hipStream_t stream) {
  const float* user_emb = (const float*)d_in[0];
  const float* item_emb = (const float*)d_in[1];
  const float* vals     = (const float*)d_in[2];
  const int*   users    = (const int*)d_in[3];
  const int*   pos      = (const int*)d_in[4];
  const int*   neg      = (const int*)d_in[5];
  const int*   rows     = (const int*)d_in[6];
  const int*   cols     = (const int*)d_in[7];

  const size_t EMB_N = (size_t)N_NODES_C * EMB_DIM_C;  // 9,600,000 floats
  float* embA      = (float*)d_ws;
  float* embB      = embA + EMB_N;
  float* accB      = embB + EMB_N;
  float* csr_val   = accB + EMB_N;
  int*   csr_col   = (int*)(csr_val + NNZ_C);
  int*   row_start = csr_col + NNZ_C;
  int*   cursor    = row_start + (N_NODES_C + 1);
  int*   counts    = cursor + N_NODES_C;
  float* partials  = (float*)(counts + N_NODES_C);

  const int init_blocks = (N_NODES_C * EMB_DIM_C + 255) / 256;  // 37500
  lg_init<<<init_blocks, 256, 0, stream>>>(user_emb, item_emb, embA, accB, counts);
  lg_count<<<2048, 256, 0, stream>>>(rows, counts);
  lg_scan<<<1, 1024, 0, stream>>>(counts, row_start, cursor);
  lg_scatter<<<2048, 256, 0, stream>>>(rows, cols, vals, cursor, csr_col, csr_val);

  const int spmm_blocks = (N_NODES_C * 32 + 255) / 256;  // 18750 (1 wave/row)
  lg_spmm<<<spmm_blocks, 256, 0, stream>>>(embA, row_start, csr_col, csr_val, embB, accB);
  lg_spmm<<<spmm_blocks, 256, 0, stream>>>(embB, row_start, csr_col, csr_val, embA, accB);
  lg_spmm<<<spmm_blocks, 256, 0, stream>>>(embA, row_start, csr_col, csr_val, embB, accB);

  const int score_blocks = (BATCH_C / 16) * 32 / 256;  // 128 blocks, 1024 waves
  lg_score<<<score_blocks, 256, 0, stream>>>(accB, users, pos, neg, partials);
  lg_reduce<<<1, 256, 0, stream>>>(partials, (float*)d_out);
}